// SCloss_8486855377005
// MI455X (gfx1250) — compile-verified
//
#include <hip/hip_runtime.h>
#include <hip/hip_bf16.h>

#define TILE 32
#define HALO 2
#define SW   (TILE + 2 * HALO)   // 36
#define SP   (SW + 1)            // 37, padded stride vs 64-bank LDS

typedef float v2f __attribute__((ext_vector_type(2)));
typedef float v8f __attribute__((ext_vector_type(8)));

// Fast TRANS-pipe math: v_exp_f32 / v_log_f32 are base-2 on CDNA5.
__device__ __forceinline__ float fexp(float x) { return __builtin_amdgcn_exp2f(x * 1.4426950408889634f); }
__device__ __forceinline__ float flog(float x) { return 0.6931471805599453f * __builtin_amdgcn_logf(x); }
__device__ __forceinline__ float frcp(float x) { return __builtin_amdgcn_rcpf(x); }
__device__ __forceinline__ float fsig(float x) { return frcp(1.0f + fexp(-x)); }

// Wave32 reduction via V_WMMA_F32_16X16X4_F32:
// A(16x4) holds one partial per lane (A[m][0]=lane m, A[m][2]=lane 16+m, rest 0),
// B(4x16) = ones  ->  D[m][n] = acc_m + acc_{16+m} for every n.
// Summing D's 8 VGPRs per lane leaves S_lo in lanes 0-15 and S_hi in lanes 16-31.
__device__ __forceinline__ float wave_reduce_wmma(float acc) {
    v2f a; a[0] = acc;  a[1] = 0.0f;
    v2f b; b[0] = 1.0f; b[1] = 1.0f;
    v8f c = {};
    v8f d = __builtin_amdgcn_wmma_f32_16x16x4_f32(false, a, false, b, (short)0, c, false, false);
    float s = d[0] + d[1] + d[2] + d[3] + d[4] + d[5] + d[6] + d[7];
    return __shfl(s, 0, 32) + __shfl(s, 16, 32);
}

__global__ void __launch_bounds__(256)
sc_loss_partial(const float* __restrict__ cmap, const int* __restrict__ tgt,
                float* __restrict__ part, int B, int H, int W) {
    __shared__ float ss[SW * SP];   // sigmoid(c_map) tile with halo
    __shared__ float st[SW * SP];   // target tile (0/1), OOB encoded as 2.0
    __shared__ float wred[8];

    static const int DX[8] = { 1, 0, -1, 1, -1, 1, 0, -1 };
    static const int DY[8] = { 1, 1,  1, 0,  0, -1, -1, -1 };

    const int tilesx   = W / TILE;
    const int tilesPer = tilesx * (H / TILE);
    const int blk = blockIdx.x;
    const int b   = blk / tilesPer;
    const int t   = blk - b * tilesPer;
    const int r0  = (t / tilesx) * TILE;
    const int c0  = (t - (t / tilesx) * tilesx) * TILE;
    const float* cimg = cmap + (size_t)b * H * W;
    const int*   timg = tgt  + (size_t)b * H * W;

    // ---- stage halo tile into LDS (sigmoid computed once per pixel here) ----
    for (int idx = threadIdx.x; idx < SW * SW; idx += blockDim.x) {
        int lr = idx / SW, lc = idx - lr * SW;
        int gr = r0 - HALO + lr, gc = c0 - HALO + lc;
        float sv = 0.0f, tv = 2.0f;   // OOB: mask-killing sentinel
        if (gr >= 0 && gr < H && gc >= 0 && gc < W) {
            size_t g = (size_t)gr * W + gc;
            sv = fsig(cimg[g]);
            tv = (float)timg[g];
        }
        ss[lr * SP + lc] = sv;
        st[lr * SP + lc] = tv;
    }
    __syncthreads();

    // ---- per-pixel fused loss: 8 dirs x K in {1,2}, unified-select formula ----
    float acc = 0.0f;
    const int lx  = threadIdx.x & 31;
    const int ly0 = threadIdx.x >> 5;   // 8 waves -> rows ly0, ly0+8, +16, +24
#pragma unroll
    for (int sub = 0; sub < 4; ++sub) {
        const int ly = ly0 + sub * 8;
        const int ci = (ly + HALO) * SP + (lx + HALO);
        const float sc = ss[ci];
        const float tc = st[ci];
#pragma unroll
        for (int i = 0; i < 8; ++i) {
            const int dy = DY[i], dx = DX[i];
#pragma unroll
            for (int k = 1; k <= 2; ++k) {
                const int ni = (ly + HALO - k * dy) * SP + (lx + HALO - k * dx);
                const float sn = ss[ni];
                const float tn = st[ni];
                // mutually exclusive masks (tn==2.0 => all false => w=0)
                const bool cpos = (tc > 0.5f) & (tn > 0.5f) & (tn < 1.5f);
                const bool cneg = (tc < 0.5f) & (tn < 0.5f);
                const bool cnb  = (tc > 0.5f) & (tn < 0.5f);
                const float wn = (k == 1) ? 0.7f : 0.35f;   // 0.7*(n1 + 0.5*n2)
                const float wp = (k == 1) ? 0.3f : 0.15f;   // 0.3*(p1 + 0.5*p2)
                const float w  = cneg ? wn : (cpos ? wp : (cnb ? 0.35f : 0.0f));
                const float pp  = sc * sn;
                const float la1 = cpos ? (sn + 1e-4f) : (1.0f - sn + 1e-4f);
                const float la2 = cneg ? (1.0f - sc + 1e-4f)
                                       : (cnb ? (sc + 1e-5f) : (sc + 1e-4f));
                const float la3 = cpos ? (pp + 1e-4f) : (1.0f - pp + 1e-4f);
                const float ea  = cnb ? -sn : -pp;
                const float cA  = cnb ? 0.8f : 1.0f;
                const float cB  = cnb ? 0.2f : 1.0f;
                const float num = -(cA * flog(la1) + cB * flog(la2));
                const float den = 1e-5f + fexp(ea) - flog(la3);
                acc += w * num * frcp(den);
            }
        }
    }

    // ---- WMMA wave reduction + LDS block reduction (no divergence: EXEC all-1) ----
    const float wtot = wave_reduce_wmma(acc);
    if ((threadIdx.x & 31) == 0) wred[threadIdx.x >> 5] = wtot;
    __syncthreads();
    if (threadIdx.x == 0) {
        float s = 0.0f;
#pragma unroll
        for (int i = 0; i < 8; ++i) s += wred[i];
        part[blockIdx.x] = s;
    }
}

__global__ void __launch_bounds__(256)
sc_loss_final(const float* __restrict__ part, int n, float* __restrict__ out, float scale) {
    __shared__ float wred[8];
    float acc = 0.0f;
    for (int i = threadIdx.x; i < n; i += 256) acc += part[i];
    const float wtot = wave_reduce_wmma(acc);
    if ((threadIdx.x & 31) == 0) wred[threadIdx.x >> 5] = wtot;
    __syncthreads();
    if (threadIdx.x == 0) {
        float s = 0.0f;
#pragma unroll
        for (int i = 0; i < 8; ++i) s += wred[i];
        out[0] = s * scale;
    }
}

extern "C" void kernel_launch(void* const* d_in, const int* in_sizes, int n_in,
                              void* d_out, int out_size, void* d_ws, size_t ws_size,
                              hipStream_t stream) {
    const float* cmap = (const float*)d_in[0];
    const int*   tgt  = (const int*)d_in[1];
    const int H = 512, W = 512;
    const int B = in_sizes[0] / (H * W);            // (B,1,H,W) flat
    const int tiles = (H / TILE) * (W / TILE) * B;  // 2048 for B=8
    float* part = (float*)d_ws;

    sc_loss_partial<<<tiles, 256, 0, stream>>>(cmap, tgt, part, B, H, W);

    const float scale = 1.0f / ((float)B * 8.0f * (float)H * (float)W); // mean over (B,8,H,W)
    sc_loss_final<<<1, 256, 0, stream>>>(part, tiles, (float*)d_out, scale);
}